// SimpleSSM_55344948576862
// MI455X (gfx1250) — compile-verified
//
#include <hip/hip_runtime.h>

// ---------------------------------------------------------------------------
// SimpleSSM on MI455X (gfx1250): blocked parallel scan, all-f32 WMMA.
//   x:(16,8192,128) h0:(16,128) A,B,C,D:(128,128)
//   h_t = h_{t-1} A^T + u_t B^T ; y_t = h_t C^T + u_t D^T
// Workspace requirement: ~66.1 MB  (XB/H 64MB + l_end 1MB + h_start 1MB + A^64 64KB)
// ---------------------------------------------------------------------------

typedef float v2f __attribute__((ext_vector_type(2)));
typedef float v8f __attribute__((ext_vector_type(8)));

#define WMMA4(a, b, c) \
  __builtin_amdgcn_wmma_f32_16x16x4_f32(false, (a), false, (b), (short)0, (c), false, false)

#define BS   16            // batch == WMMA M
#define TT   8192          // time steps
#define DD   128           // d_in == d_h == d_out
#define LCH  64            // chunk length
#define NCH  128           // number of chunks (TT/LCH)

// ---------------------------------------------------------------------------
// Kernel 1: XB = x @ B^T  -> ws ;  XD = x @ D^T -> y buffer (pre-accumulate).
// Block = one t (8 waves = 8 n-tiles). XB layout: (T, 16, 128) time-major.
// ---------------------------------------------------------------------------
__global__ void k_gemm_xbd(const float* __restrict__ x, const float* __restrict__ Bw,
                           const float* __restrict__ Dw, float* __restrict__ xb,
                           float* __restrict__ y) {
  const int t    = blockIdx.x;
  const int nt   = threadIdx.x >> 5;
  const int lane = threadIdx.x & 31;
  const int nlo  = lane & 15;
  const int khi  = lane >> 4;

  v8f accB = (v8f)0.0f, accD = (v8f)0.0f;
  const float* xrow = x  + (size_t)nlo * (TT * DD) + (size_t)t * DD;  // A-frag: m = batch
  const float* brow = Bw + (size_t)(nt * 16 + nlo) * DD;              // weight form W[n][k]
  const float* drow = Dw + (size_t)(nt * 16 + nlo) * DD;
#pragma unroll
  for (int g = 0; g < 32; ++g) {
    const int k0 = 4 * g + 2 * khi;
    v2f a  = *(const v2f*)(xrow + k0);
    v2f bb = *(const v2f*)(brow + k0);
    v2f bd = *(const v2f*)(drow + k0);
    accB = WMMA4(a, bb, accB);
    accD = WMMA4(a, bd, accD);
  }
  float* xbo = xb + (size_t)t * (BS * DD) + nt * 16 + nlo;
  float* yo  = y  + (size_t)t * DD + nt * 16 + nlo;
#pragma unroll
  for (int r = 0; r < 8; ++r) {
    const int row = r + 8 * khi;
    xbo[row * DD] = accB[r];                     // XB[t][b][n]
    yo[(size_t)row * (TT * DD)] = accD[r];       // y[b][t][n]  (holds XD for now)
  }
}

// ---------------------------------------------------------------------------
// Kernel 2: A^64 by 6 in-place squarings of a 128x128 matrix in LDS (64 KB).
// One block, 8 waves; wave w owns output row-tile m = w.
// ---------------------------------------------------------------------------
__global__ void k_apow(const float* __restrict__ Aw, float* __restrict__ apow) {
  extern __shared__ float P[];  // 128*128 floats
  const int tid  = threadIdx.x;                 // 256
  const int wv   = tid >> 5;
  const int lane = tid & 31;
  const int nlo  = lane & 15;
  const int khi  = lane >> 4;

  for (int i = tid; i < DD * DD; i += 256) P[i] = Aw[i];
  __syncthreads();

  for (int it = 0; it < 6; ++it) {              // A^2, A^4, ... A^64
    v8f acc[8];
    const int mbase = wv * 16;
#pragma unroll
    for (int nt = 0; nt < 8; ++nt) {
      v8f c = (v8f)0.0f;
#pragma unroll 8
      for (int g = 0; g < 32; ++g) {
        const int k0 = 4 * g + 2 * khi;
        v2f a = *(const v2f*)&P[(mbase + nlo) * DD + k0];   // P[m][k0..k0+1]
        v2f b;
        b.x = P[(k0)     * DD + nt * 16 + nlo];              // P[k][n]
        b.y = P[(k0 + 1) * DD + nt * 16 + nlo];
        c = WMMA4(a, b, c);
      }
      acc[nt] = c;
    }
    __syncthreads();                            // all reads done -> safe in-place
#pragma unroll
    for (int nt = 0; nt < 8; ++nt)
#pragma unroll
      for (int r = 0; r < 8; ++r)
        P[(mbase + r + 8 * khi) * DD + nt * 16 + nlo] = acc[nt][r];
    __syncthreads();
  }
  for (int i = tid; i < DD * DD; i += 256) apow[i] = P[i];
}

// ---------------------------------------------------------------------------
// Kernels 3 & 5: per-chunk scan, one wave per chunk, full 16x128 h in regs.
// REPLAY=0: h starts at 0; final h -> lend[c].
// REPLAY=1: h starts at hstart[c]; every step's h overwrites XB[t] (-> H).
// LDS: 64 KB shared A-weight + 8 KB h relayout stage per wave (4 waves/block).
// ---------------------------------------------------------------------------
template <int REPLAY>
__global__ void k_scan(const float* __restrict__ Aw, const float* __restrict__ hstart,
                       float* __restrict__ xb, float* __restrict__ lend) {
  extern __shared__ float smem[];
  float* As = smem;                              // 16384 floats
  const int tid  = threadIdx.x;                  // 128 (4 waves)
  const int wv   = tid >> 5;
  const int lane = tid & 31;
  const int nlo  = lane & 15;
  const int khi  = lane >> 4;
  float* hstage = smem + DD * DD + wv * (BS * DD);

  for (int i = tid; i < DD * DD; i += 128) As[i] = Aw[i];
  __syncthreads();

  const int c = blockIdx.x * 4 + wv;             // chunk id

  v8f h[8];
  if (REPLAY) {
    const float* hs = hstart + (size_t)c * (BS * DD) + nlo;
#pragma unroll
    for (int nt = 0; nt < 8; ++nt)
#pragma unroll
      for (int r = 0; r < 8; ++r)
        h[nt][r] = hs[(r + 8 * khi) * DD + nt * 16];
  } else {
#pragma unroll
    for (int nt = 0; nt < 8; ++nt) h[nt] = (v8f)0.0f;
  }

  const int t0 = c * LCH;
  for (int j = 0; j < LCH; ++j) {
    const int t = t0 + j;
    // stage h in C-layout (per-wave private slice; DS ops are in-order per wave)
#pragma unroll
    for (int nt = 0; nt < 8; ++nt)
#pragma unroll
      for (int r = 0; r < 8; ++r)
        hstage[(r + 8 * khi) * DD + nt * 16 + nlo] = h[nt][r];
    // accumulator starts as XB[t] tile
    float* xbt = xb + (size_t)t * (BS * DD);
#pragma unroll
    for (int nt = 0; nt < 8; ++nt)
#pragma unroll
      for (int r = 0; r < 8; ++r)
        h[nt][r] = xbt[(r + 8 * khi) * DD + nt * 16 + nlo];
    // h = stage @ A^T + h   (A-frags re-read from stage in A-layout)
#pragma unroll 4
    for (int g = 0; g < 32; ++g) {
      const int k0 = 4 * g + 2 * khi;
      v2f a = *(const v2f*)&hstage[nlo * DD + k0];
#pragma unroll
      for (int nt = 0; nt < 8; ++nt) {
        v2f b = *(const v2f*)&As[(nt * 16 + nlo) * DD + k0];
        h[nt] = WMMA4(a, b, h[nt]);
      }
    }
    if (REPLAY) {                                // H[t] overwrites XB[t]
#pragma unroll
      for (int nt = 0; nt < 8; ++nt)
#pragma unroll
        for (int r = 0; r < 8; ++r)
          xbt[(r + 8 * khi) * DD + nt * 16 + nlo] = h[nt][r];
    }
  }
  if (!REPLAY) {
    float* lo = lend + (size_t)c * (BS * DD) + nlo;
#pragma unroll
    for (int nt = 0; nt < 8; ++nt)
#pragma unroll
      for (int r = 0; r < 8; ++r)
        lo[(r + 8 * khi) * DD + nt * 16] = h[nt][r];
  }
}

// ---------------------------------------------------------------------------
// Kernel 4: serial chunk-level scan: h <- h @ (A^64)^T + lend[c]. One block,
// 8 waves (one n-tile each), h exchanged through an LDS stage each chunk.
// Emits hstart[c] (state BEFORE chunk c) and h_final.
// ---------------------------------------------------------------------------
__global__ void k_chunkscan(const float* __restrict__ apow, const float* __restrict__ h0,
                            const float* __restrict__ lend, float* __restrict__ hstart,
                            float* __restrict__ hfinal) {
  extern __shared__ float smem[];
  float* As    = smem;                 // 16384 floats
  float* stage = smem + DD * DD;       // 2048 floats
  const int tid  = threadIdx.x;        // 256
  const int nt   = tid >> 5;
  const int lane = tid & 31;
  const int nlo  = lane & 15;
  const int khi  = lane >> 4;

  for (int i = tid; i < DD * DD; i += 256) As[i] = apow[i];
  for (int i = tid; i < BS * DD; i += 256) stage[i] = h0[i];
  __syncthreads();

  for (int c = 0; c < NCH; ++c) {
    for (int i = tid; i < BS * DD; i += 256) hstart[(size_t)c * (BS * DD) + i] = stage[i];
    v8f acc;
    const float* le = lend + (size_t)c * (BS * DD) + nt * 16 + nlo;
#pragma unroll
    for (int r = 0; r < 8; ++r) acc[r] = le[(r + 8 * khi) * DD];
#pragma unroll 4
    for (int g = 0; g < 32; ++g) {
      const int k0 = 4 * g + 2 * khi;
      v2f a = *(const v2f*)&stage[nlo * DD + k0];
      v2f b = *(const v2f*)&As[(nt * 16 + nlo) * DD + k0];
      acc = WMMA4(a, b, acc);
    }
    __syncthreads();                   // all reads of stage complete
#pragma unroll
    for (int r = 0; r < 8; ++r)
      stage[(r + 8 * khi) * DD + nt * 16 + nlo] = acc[r];
    __syncthreads();
  }
  for (int i = tid; i < BS * DD; i += 256) hfinal[i] = stage[i];
}

// ---------------------------------------------------------------------------
// Kernel 6: y = H @ C^T + y(=XD).  Block = one t, 8 waves = 8 n-tiles.
// ---------------------------------------------------------------------------
__global__ void k_gemm_y(const float* __restrict__ H, const float* __restrict__ Cw,
                         float* __restrict__ y) {
  const int t    = blockIdx.x;
  const int nt   = threadIdx.x >> 5;
  const int lane = threadIdx.x & 31;
  const int nlo  = lane & 15;
  const int khi  = lane >> 4;

  float* yo = y + (size_t)t * DD + nt * 16 + nlo;
  v8f acc;
#pragma unroll
  for (int r = 0; r < 8; ++r) acc[r] = yo[(size_t)(r + 8 * khi) * (TT * DD)];
  const float* hrow = H  + (size_t)t * (BS * DD) + nlo * DD;  // H[t][m][k]
  const float* crow = Cw + (size_t)(nt * 16 + nlo) * DD;
#pragma unroll
  for (int g = 0; g < 32; ++g) {
    const int k0 = 4 * g + 2 * khi;
    v2f a = *(const v2f*)(hrow + k0);
    v2f b = *(const v2f*)(crow + k0);
    acc = WMMA4(a, b, acc);
  }
#pragma unroll
  for (int r = 0; r < 8; ++r) yo[(size_t)(r + 8 * khi) * (TT * DD)] = acc[r];
}

// ---------------------------------------------------------------------------
extern "C" void kernel_launch(void* const* d_in, const int* in_sizes, int n_in,
                              void* d_out, int out_size, void* d_ws, size_t ws_size,
                              hipStream_t stream) {
  (void)in_sizes; (void)n_in; (void)out_size; (void)ws_size;
  const float* x  = (const float*)d_in[0];
  const float* h0 = (const float*)d_in[1];
  const float* A  = (const float*)d_in[2];
  const float* B  = (const float*)d_in[3];
  const float* C  = (const float*)d_in[4];
  const float* D  = (const float*)d_in[5];

  float* y      = (float*)d_out;                       // (16,8192,128)
  float* hfinal = y + (size_t)BS * TT * DD;            // (16,128)

  float* XB     = (float*)d_ws;                        // (T,16,128) -> becomes H
  float* LEND   = XB + (size_t)TT * BS * DD;           // (NCH,16,128)
  float* HSTART = LEND + (size_t)NCH * BS * DD;        // (NCH,16,128)
  float* APOW   = HSTART + (size_t)NCH * BS * DD;      // (128,128)

  const size_t LDS_A    = (size_t)DD * DD * sizeof(float);      // 64 KB
  const size_t LDS_SCAN = LDS_A + 4 * (size_t)BS * DD * sizeof(float);   // 96 KB
  const size_t LDS_CHNK = LDS_A + (size_t)BS * DD * sizeof(float);       // 72 KB

  k_gemm_xbd<<<TT, 256, 0, stream>>>(x, B, D, XB, y);            // XB, XD
  k_apow<<<1, 256, LDS_A, stream>>>(A, APOW);                    // A^64
  k_scan<0><<<NCH / 4, 128, LDS_SCAN, stream>>>(A, nullptr, XB, LEND);   // local scans
  k_chunkscan<<<1, 256, LDS_CHNK, stream>>>(APOW, h0, LEND, HSTART, hfinal);
  k_scan<1><<<NCH / 4, 128, LDS_SCAN, stream>>>(A, HSTART, XB, nullptr); // replay -> H
  k_gemm_y<<<TT, 256, 0, stream>>>(XB, C, y);                    // y = H C^T + XD
}